// KinematicDeltaNet_30288109371506
// MI455X (gfx1250) — compile-verified
//
#include <hip/hip_runtime.h>
#include <math.h>

typedef __attribute__((ext_vector_type(2))) float v2f;
typedef __attribute__((ext_vector_type(8))) float v8f;
typedef unsigned int u32x4 __attribute__((ext_vector_type(4)));
typedef int i32x4 __attribute__((ext_vector_type(4)));
typedef int i32x8 __attribute__((ext_vector_type(8)));

#define BATCH  512
#define SEQ    120
#define DIM    64
#define MROWS  (BATCH * SEQ)   /* 61440 */
#define NBANDS 8
#define WROW   65               /* LDS row stride in dwords (TDM-padded)      */
#define WMAT   4160             /* 64 rows * 65 dwords per staged 64x64 matrix */

// Workspace layout (float offsets). Total = 15,794,176 floats ~= 63.2 MB.
#define DFT_OFF 0
#define GAM_OFF 4096
#define Q_OFF   65536
#define K_OFF   (Q_OFF + MROWS * DIM)
#define V_OFF   (K_OFF + MROWS * DIM)
#define O_OFF   (V_OFF + MROWS * DIM)

// ---------------------------------------------------------------------------
// TDM: async-DMA one 64x64 f32 matrix (row-major, stride 64) from global into
// LDS at byte offset lds_off, inserting 1 dword of padding every 64 dwords
// (=> 65-dword LDS row stride, bank-conflict free). Tracked by TENSORcnt.
// D# layout per CDNA5 ISA ch.8 (group0 128b, group1 256b, groups 2/3 unused).
// ---------------------------------------------------------------------------
__device__ __forceinline__ void tdm_load_w64(const float* gptr, unsigned lds_off) {
  unsigned long long ga = (unsigned long long)(uintptr_t)gptr;
  u32x4 g0;
  g0.x = 1u;                                         // count=1 valid descriptor
  g0.y = lds_off;                                    // lds_addr (bytes)
  g0.z = (unsigned)(ga & 0xFFFFFFFFu);               // global_addr[31:0]
  g0.w = (unsigned)((ga >> 32) & 0x01FFFFFFu)        // global_addr[56:32]
       | (2u << 30);                                 // type=2 ("image")
  i32x8 g1;
  g1[0] = (2 << 16)                                  // data_size = 4 bytes
        | (1 << 20)                                  // pad_enable
        | (5 << 22)                                  // pad_interval: 64 dwords
        | (0 << 25);                                 // pad_amount: 1 dword
  g1[1] = (64 << 16);                                // tensor_dim0 = 64
  g1[2] = (64 << 16);                                // tensor_dim1 = 64 (lo16)
  g1[3] = (64 << 16);                                // tile_dim0 = 64
  g1[4] = 64;                                        // tile_dim1 = 64
  g1[5] = 64;                                        // tensor_dim0_stride = 64
  g1[6] = 0;
  g1[7] = 0;
  i32x4 z4 = {0, 0, 0, 0};
#if defined(__clang_major__) && (__clang_major__ >= 23)
  i32x8 z8 = {0, 0, 0, 0, 0, 0, 0, 0};
  __builtin_amdgcn_tensor_load_to_lds(g0, g1, z4, z4, z8, 0);
#else
  __builtin_amdgcn_tensor_load_to_lds(g0, g1, z4, z4, 0);
#endif
}

__device__ __forceinline__ unsigned lds_offset_of(const void* p) {
  // flat LDS aperture address: low 32 bits are the LDS byte offset
  return (unsigned)(uintptr_t)p;
}

// ---------------------------------------------------------------------------
// Kernel 0: build 64x64 real-DFT matrix. Column f (0..31) = cos(2*pi*f*d/64),
// column 32+f = -sin(2*pi*f*d/64). h @ DFT gives [Re X | Im X] of rfft(h).
// ---------------------------------------------------------------------------
__global__ void build_dft(float* __restrict__ dft) {
  int d = threadIdx.x;  // 64 threads
  const float w = 6.283185307179586476925f / 64.0f;
  for (int f = 0; f < 32; ++f) {
    float ang = w * (float)(f * d);
    dft[d * 64 + f]      =  cosf(ang);
    dft[d * 64 + 32 + f] = -sinf(ang);
  }
}

// ---------------------------------------------------------------------------
// Kernel 1: zc_rms -> {q (l2norm), k (l2norm), v (+bv), gamma (DFT+MLP)}
// One block = 64 token rows. Weights TDM-prefetched to LDS while zc_rms runs.
// 8 waves x 2 WMMA f32 16x16x4 tiles per weight matrix.
// ---------------------------------------------------------------------------
__global__ __launch_bounds__(256)
void prep_kernel(const float* __restrict__ x,  const float* __restrict__ g_norm,
                 const float* __restrict__ Wq, const float* __restrict__ Wk,
                 const float* __restrict__ Wv, const float* __restrict__ bv,
                 const float* __restrict__ W1, const float* __restrict__ b1,
                 const float* __restrict__ W2, const float* __restrict__ b2,
                 const float* __restrict__ dft, float* __restrict__ ws) {
  __shared__ float hts[64][65];   // normalized activations
  __shared__ float stg[64][65];   // staged GEMM result (reused per weight matrix)
  __shared__ float rsc[64];       // per-row l2norm scales
  __shared__ float wlds[4][WMAT]; // Wq|Wk|Wv|DFT, 65-dword row stride via TDM pad

  const int  tid  = threadIdx.x;
  const int  row  = tid >> 2;           // local token row 0..63
  const int  c0   = (tid & 3) << 4;     // 16-col slice
  const long g0   = (long)blockIdx.x * 64;
  const long grow = g0 + row;
  const int  wv   = tid >> 5;           // wave id 0..7

  // ---- wave 0 kicks off the 4 async weight DMAs (EXEC-independent) ----
  if (wv == 0) {
    tdm_load_w64(Wq,  lds_offset_of(&wlds[0][0]));
    tdm_load_w64(Wk,  lds_offset_of(&wlds[1][0]));
    tdm_load_w64(Wv,  lds_offset_of(&wlds[2][0]));
    tdm_load_w64(dft, lds_offset_of(&wlds[3][0]));
  }

  // ---- zero-centred RMS norm of x tile (4 lanes cooperate per row) ----
  {
    const float* xp = x + grow * DIM + c0;
    float xv[16], s = 0.f, ss = 0.f;
#pragma unroll
    for (int j = 0; j < 16; ++j) { float t = xp[j]; xv[j] = t; s += t; ss += t * t; }
    s  += __shfl_xor(s, 1, 32);  s  += __shfl_xor(s, 2, 32);
    ss += __shfl_xor(ss, 1, 32); ss += __shfl_xor(ss, 2, 32);
    float mu  = s * (1.0f / 64.0f);
    float var = ss * (1.0f / 64.0f) - mu * mu;
    float sc  = rsqrtf(var + 1e-8f);
#pragma unroll
    for (int j = 0; j < 16; ++j) hts[row][c0 + j] = (xv[j] - mu) * sc * g_norm[c0 + j];
  }
  if (wv == 0) __builtin_amdgcn_s_wait_tensorcnt(0);  // weight DMAs landed
  __syncthreads();

  const int lane = tid & 31;
  const int half = lane >> 4;      // 0 | 1
  const int lcol = lane & 15;      // M (for A) / N (for B,C)
  const int koff = half << 1;      // K sub: lanes 0-15 -> K+0,1; 16-31 -> K+2,3

  for (int m = 0; m < 4; ++m) {
    const float* W = &wlds[m][0];
    // Each wave computes tiles 2*wv and 2*wv+1 (ti -> rows (ti>>2)*16, cols (ti&3)*16)
#pragma unroll
    for (int tix = 0; tix < 2; ++tix) {
      const int ti = wv * 2 + tix;
      const int r0 = (ti >> 2) << 4;
      const int n0 = (ti & 3) << 4;
      v8f acc = {};
#pragma unroll
      for (int k0 = 0; k0 < 64; k0 += 4) {
        v2f a, bfrag;
        a.x     = hts[r0 + lcol][k0 + koff];
        a.y     = hts[r0 + lcol][k0 + koff + 1];
        bfrag.x = W[(k0 + koff) * WROW + n0 + lcol];
        bfrag.y = W[(k0 + koff + 1) * WROW + n0 + lcol];
        acc = __builtin_amdgcn_wmma_f32_16x16x4_f32(false, a, false, bfrag,
                                                    (short)0, acc, false, false);
      }
#pragma unroll
      for (int i = 0; i < 8; ++i)
        stg[r0 + i + (half << 3)][n0 + lcol] = acc[i];
    }
    __syncthreads();

    if (m <= 1) {                       // q or k : per-row l2 normalize
      if (tid < 64) {
        float sq = 0.f;
        for (int c = 0; c < 64; ++c) { float t = stg[tid][c]; sq += t * t; }
        rsc[tid] = rsqrtf(sq + 1e-8f);
      }
      __syncthreads();
      float* dst = ws + (m == 0 ? Q_OFF : K_OFF) + grow * DIM + c0;
      float  rs  = rsc[row];
#pragma unroll
      for (int j = 0; j < 16; ++j) dst[j] = stg[row][c0 + j] * rs;
    } else if (m == 2) {                // v : add bias
      float* dst = ws + V_OFF + grow * DIM + c0;
#pragma unroll
      for (int j = 0; j < 16; ++j) dst[j] = stg[row][c0 + j] + bv[c0 + j];
    } else {                            // DFT -> band energies -> MLP -> gamma
      if (tid < 64) {
        float band[NBANDS];
#pragma unroll
        for (int i = 0; i < NBANDS; ++i) band[i] = 0.f;
        for (int f = 0; f < 32; ++f) {  // bin 32 dropped (33//8 = 4 per band)
          float cr = stg[tid][f], ci = stg[tid][32 + f];
          band[f >> 2] += sqrtf(cr * cr + ci * ci) * 0.25f;
        }
        float tot = 0.f;
        for (int i = 0; i < NBANDS; ++i) tot += band[i];
        float inv = 1.0f / fmaxf(tot, 1e-8f);
        float outz = b2[0];
        for (int h = 0; h < 16; ++h) {
          float z = b1[h];
          for (int i = 0; i < NBANDS; ++i) z += band[i] * inv * W1[i * 16 + h];
          float sig = 1.0f / (1.0f + __expf(-z));
          outz += (z * sig) * W2[h];   // silu
        }
        float raw = 1.0f / (1.0f + __expf(-outz));
        ws[GAM_OFF + g0 + tid] = 0.5f + 0.49f * raw;
      }
    }
    __syncthreads();
  }
}

// ---------------------------------------------------------------------------
// Kernel 2: delta-rule scan. One block per batch; S (64x64) in registers,
// thread owns row tid>>2, cols (tid&3)*16..+15. Reductions via shfl_xor.
// ---------------------------------------------------------------------------
__global__ __launch_bounds__(256)
void scan_kernel(const float* __restrict__ wq, const float* __restrict__ wk,
                 const float* __restrict__ wv, const float* __restrict__ wg,
                 float* __restrict__ wo) {
  const int  b    = blockIdx.x;
  const int  tid  = threadIdx.x;
  const int  r    = tid >> 2;
  const int  c0   = (tid & 3) << 4;
  const long base = (long)b * SEQ * DIM;

  float S[16];
#pragma unroll
  for (int j = 0; j < 16; ++j) S[j] = 0.f;

  float kt[16], qt[16], vr, gt;
  {
    const float4* kp = (const float4*)(wk + base + c0);
    const float4* qp = (const float4*)(wq + base + c0);
#pragma unroll
    for (int j = 0; j < 4; ++j) {
      float4 a = kp[j]; kt[4*j] = a.x; kt[4*j+1] = a.y; kt[4*j+2] = a.z; kt[4*j+3] = a.w;
      float4 c = qp[j]; qt[4*j] = c.x; qt[4*j+1] = c.y; qt[4*j+2] = c.z; qt[4*j+3] = c.w;
    }
    vr = wv[base + r];
    gt = wg[b * SEQ];
  }

  for (int t = 0; t < SEQ; ++t) {
    float nk[16], nq[16], nvr = 0.f, ngt = 0.f;
    if (t + 1 < SEQ) {                                   // software pipeline
      const long off = base + (long)(t + 1) * DIM;
      const float4* kp = (const float4*)(wk + off + c0);
      const float4* qp = (const float4*)(wq + off + c0);
#pragma unroll
      for (int j = 0; j < 4; ++j) {
        float4 a = kp[j]; nk[4*j] = a.x; nk[4*j+1] = a.y; nk[4*j+2] = a.z; nk[4*j+3] = a.w;
        float4 c = qp[j]; nq[4*j] = c.x; nq[4*j+1] = c.y; nq[4*j+2] = c.z; nq[4*j+3] = c.w;
      }
      nvr = wv[off + r];
      ngt = wg[b * SEQ + t + 1];
    } else {
#pragma unroll
      for (int j = 0; j < 16; ++j) { nk[j] = 0.f; nq[j] = 0.f; }
    }
    if (t + 2 < SEQ) {                                   // global_prefetch_b8
      __builtin_prefetch(wk + base + (long)(t + 2) * DIM + c0, 0, 0);
      __builtin_prefetch(wq + base + (long)(t + 2) * DIM + c0, 0, 0);
    }

    float pred = 0.f;
#pragma unroll
    for (int j = 0; j < 16; ++j) pred += S[j] * kt[j];
    pred += __shfl_xor(pred, 1, 32);
    pred += __shfl_xor(pred, 2, 32);

    float delta = fminf(fmaxf(vr - pred, -5.0f), 5.0f);

    float osum = 0.f;
#pragma unroll
    for (int j = 0; j < 16; ++j) {
      float sn = gt * S[j] + delta * kt[j];
      sn = fminf(fmaxf(sn, -10.0f), 10.0f);
      S[j] = sn;
      osum += sn * qt[j];
    }
    osum += __shfl_xor(osum, 1, 32);
    osum += __shfl_xor(osum, 2, 32);
    if ((tid & 3) == 0) wo[base + (long)t * DIM + r] = osum;

#pragma unroll
    for (int j = 0; j < 16; ++j) { kt[j] = nk[j]; qt[j] = nq[j]; }
    vr = nvr; gt = ngt;
  }
}

// ---------------------------------------------------------------------------
// Kernel 3: out = x + zc_rms(o, g_post) @ Wo + bo   (WMMA f32 16x16x4)
// Wo is TDM-prefetched into LDS while zc_rms runs.
// ---------------------------------------------------------------------------
__global__ __launch_bounds__(256)
void out_kernel(const float* __restrict__ x,  const float* __restrict__ g_post,
                const float* __restrict__ Wo, const float* __restrict__ bo,
                const float* __restrict__ wso, float* __restrict__ out) {
  __shared__ float ots[64][65];
  __shared__ float wolds[WMAT];

  const int  tid  = threadIdx.x;
  const int  row  = tid >> 2;
  const int  c0   = (tid & 3) << 4;
  const long g0   = (long)blockIdx.x * 64;
  const long grow = g0 + row;
  const int  wv   = tid >> 5;

  if (wv == 0) tdm_load_w64(Wo, lds_offset_of(&wolds[0]));

  {
    const float* op = wso + grow * DIM + c0;
    float ov[16], s = 0.f, ss = 0.f;
#pragma unroll
    for (int j = 0; j < 16; ++j) { float t = op[j]; ov[j] = t; s += t; ss += t * t; }
    s  += __shfl_xor(s, 1, 32);  s  += __shfl_xor(s, 2, 32);
    ss += __shfl_xor(ss, 1, 32); ss += __shfl_xor(ss, 2, 32);
    float mu  = s * (1.0f / 64.0f);
    float var = ss * (1.0f / 64.0f) - mu * mu;
    float sc  = rsqrtf(var + 1e-8f);
#pragma unroll
    for (int j = 0; j < 16; ++j) ots[row][c0 + j] = (ov[j] - mu) * sc * g_post[c0 + j];
  }
  if (wv == 0) __builtin_amdgcn_s_wait_tensorcnt(0);
  __syncthreads();

  const int lane = tid & 31;
  const int half = lane >> 4;
  const int lcol = lane & 15;
  const int koff = half << 1;

#pragma unroll
  for (int tix = 0; tix < 2; ++tix) {
    const int ti = wv * 2 + tix;
    const int r0 = (ti >> 2) << 4;
    const int n0 = (ti & 3) << 4;
    v8f acc = {};
#pragma unroll
    for (int k0 = 0; k0 < 64; k0 += 4) {
      v2f a, bfrag;
      a.x     = ots[r0 + lcol][k0 + koff];
      a.y     = ots[r0 + lcol][k0 + koff + 1];
      bfrag.x = wolds[(k0 + koff) * WROW + n0 + lcol];
      bfrag.y = wolds[(k0 + koff + 1) * WROW + n0 + lcol];
      acc = __builtin_amdgcn_wmma_f32_16x16x4_f32(false, a, false, bfrag,
                                                  (short)0, acc, false, false);
    }
#pragma unroll
    for (int i = 0; i < 8; ++i) {
      const int  rr = r0 + i + (half << 3);
      const long gi = (g0 + rr) * DIM + n0 + lcol;
      out[gi] = acc[i] + bo[n0 + lcol] + x[gi];   // residual
    }
  }
}

// ---------------------------------------------------------------------------
extern "C" void kernel_launch(void* const* d_in, const int* in_sizes, int n_in,
                              void* d_out, int out_size, void* d_ws, size_t ws_size,
                              hipStream_t stream) {
  const float* x      = (const float*)d_in[0];
  const float* g_norm = (const float*)d_in[1];
  const float* Wq     = (const float*)d_in[2];
  const float* Wk     = (const float*)d_in[3];
  const float* Wv     = (const float*)d_in[4];
  const float* bv     = (const float*)d_in[5];
  const float* Wo     = (const float*)d_in[6];
  const float* bo     = (const float*)d_in[7];
  const float* g_post = (const float*)d_in[8];
  const float* W1     = (const float*)d_in[9];
  const float* b1     = (const float*)d_in[10];
  const float* W2     = (const float*)d_in[11];
  const float* b2     = (const float*)d_in[12];

  float* ws  = (float*)d_ws;
  float* out = (float*)d_out;

  build_dft<<<1, 64, 0, stream>>>(ws + DFT_OFF);
  prep_kernel<<<MROWS / 64, 256, 0, stream>>>(x, g_norm, Wq, Wk, Wv, bv,
                                              W1, b1, W2, b2, ws + DFT_OFF, ws);
  scan_kernel<<<BATCH, 256, 0, stream>>>(ws + Q_OFF, ws + K_OFF, ws + V_OFF,
                                         ws + GAM_OFF, ws + O_OFF);
  out_kernel<<<MROWS / 64, 256, 0, stream>>>(x, g_post, Wo, bo, ws + O_OFF, out);
}